// Word2VecModel_12128987644221
// MI455X (gfx1250) — compile-verified
//
#include <hip/hip_runtime.h>
#include <hip/hip_bf16.h>

typedef __attribute__((ext_vector_type(2))) float v2f;
typedef __attribute__((ext_vector_type(8))) float v8f;

#define HID   300
#define NEGS  5
#define OUTC  (NEGS + 1)   // 6 output columns: [true, 5 sampled]

// One block = one tile of 16 batch rows; 6 waves per block, wave j handles
// context slot j (j==0 -> label, j>=1 -> sampled[j-1]) for all 16 rows of
// the tile via a single diagonal-extracted WMMA chain.
__global__ __launch_bounds__(192, 4)
void w2v_nce_wmma_kernel(const float* __restrict__ syn0,
                         const float* __restrict__ syn1,
                         const float* __restrict__ biases,
                         const int*   __restrict__ inputs,
                         const int*   __restrict__ labels,
                         const int*   __restrict__ sampled,
                         float*       __restrict__ out,
                         int Bn)
{
    const int tile = blockIdx.x;            // tile of 16 batch rows
    const int j    = threadIdx.x >> 5;      // context slot 0..5 (one wave each)
    const int lane = threadIdx.x & 31;
    const int t    = lane & 15;             // M index (A row) == N index (B col)
    const int half = lane >> 4;             // K-half selector for f32 fragments
    const int koff = 2 * half;              // K = k0 + koff (+0, +1) per ISA layout

    int b = tile * 16 + t;
    const bool valid = (b < Bn);
    if (!valid) b = Bn - 1;                 // clamp loads; predicate only the store

    // A row: hidden vector h_b = syn0[inputs[b]]
    const float* arow = syn0 + (size_t)inputs[b] * HID;
    // B column t: context-j vector for batch row t of this tile
    const int ctx = (j == 0) ? labels[b] : sampled[(size_t)b * NEGS + (j - 1)];
    const float* brow = syn1 + (size_t)ctx * HID;

    // D = A(16xK) * B(Kx16) + C accumulated over K in steps of 4.
    // f32 16x4 A fragment: vgpr v, lane-half h -> K = k0 + v + 2h  (2 VGPRs = v2f)
    // f32 4x16 B fragment: same K mapping, lane = N               (2 VGPRs = v2f)
    v8f acc = {0.f, 0.f, 0.f, 0.f, 0.f, 0.f, 0.f, 0.f};
    #pragma unroll 5
    for (int k0 = 0; k0 < HID; k0 += 4) {
        v2f af = *(const v2f*)(arow + k0 + koff);   // A[t][k0+2h .. +1]
        v2f bf = *(const v2f*)(brow + k0 + koff);   // B[k0+2h .. +1][t]
        acc = __builtin_amdgcn_wmma_f32_16x16x4_f32(
                  /*neg_a=*/false, af, /*neg_b=*/false, bf,
                  /*c_mod=*/(short)0, acc, /*reuse_a=*/false, /*reuse_b=*/false);
    }

    // Diagonal D[t][t]: VGPR r holds M=r (lanes 0-15) and M=r+8 (lanes 16-31),
    // N = lane&15. So row t's diagonal lives at acc[t&7] on lane t (t<8) or
    // lane t+16 (t>=8): i.e. lanes 0-7 and 24-31.
    const bool owner = (lane < 8) || (lane >= 24);
    if (owner && valid) {
        float d;
        switch (lane & 7) {
            case 0: d = acc[0]; break;
            case 1: d = acc[1]; break;
            case 2: d = acc[2]; break;
            case 3: d = acc[3]; break;
            case 4: d = acc[4]; break;
            case 5: d = acc[5]; break;
            case 6: d = acc[6]; break;
            default: d = acc[7]; break;
        }
        float logit = d + biases[ctx];
        // col 0: softplus(-logit); cols 1..5: softplus(logit)
        float x = (j == 0) ? -logit : logit;
        // numerically stable softplus: max(x,0) + log1p(exp(-|x|))
        float ce = fmaxf(x, 0.f) + log1pf(expf(-fabsf(x)));
        out[(size_t)b * OUTC + j] = ce;
    }
}

extern "C" void kernel_launch(void* const* d_in, const int* in_sizes, int n_in,
                              void* d_out, int out_size, void* d_ws, size_t ws_size,
                              hipStream_t stream)
{
    const float* syn0    = (const float*)d_in[0];
    const float* syn1    = (const float*)d_in[1];
    const float* biases  = (const float*)d_in[2];
    const int*   inputs  = (const int*)d_in[3];
    const int*   labels  = (const int*)d_in[4];
    const int*   sampled = (const int*)d_in[5];
    float*       out     = (float*)d_out;

    const int Bn    = in_sizes[3];            // 16384
    const int tiles = (Bn + 15) / 16;         // 1024 blocks
    dim3 grid(tiles), block(192);             // 6 waves: one per context slot

    hipLaunchKernelGGL(w2v_nce_wmma_kernel, grid, block, 0, stream,
                       syn0, syn1, biases, inputs, labels, sampled, out, Bn);
}